// SymmetricPositiveDefinite_22643067584720
// MI455X (gfx1250) — compile-verified
//
#include <hip/hip_runtime.h>
#include <cmath>

typedef __attribute__((ext_vector_type(2))) float v2f;
typedef __attribute__((ext_vector_type(8))) float v8f;

#define LS        76      // LDS row stride in floats (bank-conflict-free for A/B/C access)
#define TAYLOR_N  10      // Taylor order after scaling (||A|| <= 0.5 -> remainder ~1e-11)
#define THREADS   128     // 4 waves of 32; each wave owns a 16-row band

// Wave-cooperative: rows [16*wv, 16*wv+16) of Dst = scale*(Asrc*Bsrc) (+I on diagonal).
// Asrc/Bsrc/Dst are 64x64 matrices in LDS with row stride LS.
// ct is the inner dimension: A-fragment loaded once (ds_load_b64) and reused for
// 4 independent accumulator chains -> 4-way WMMA ILP, 144 DS loads per 64 WMMA.
__device__ __forceinline__ void band_matmul(const float* Asrc, const float* Bsrc,
                                            float* Dst, int wv, int lane,
                                            float scale, bool addI)
{
    const int half = lane >> 4;   // selects K pair (A/B) and M half (C/D)
    const int mn   = lane & 15;   // M for A-frag, N for B/C-frag
    const int row0 = wv << 4;

    v8f acc0 = {}, acc1 = {}, acc2 = {}, acc3 = {};
    const int abase = (row0 + mn) * LS + 2 * half;   // + k0
    const int bbase = (2 * half) * LS + mn;          // + k0*LS (+ 16*ct)

    #pragma unroll
    for (int k0 = 0; k0 < 64; k0 += 4) {
        // A 16x4 f32 frag: VGPR v, lane-half h -> K = k0 + v + 2*h, M = mn  (8B aligned pair)
        const v2f a = *(const v2f*)(Asrc + abase + k0);
        const float* bp = Bsrc + bbase + k0 * LS;
        // B 4x16 f32 frag: VGPR v, lane-half h -> K = k0 + v + 2*h, N = 16*ct + mn
        v2f b0, b1, b2, b3;
        b0.x = bp[0];  b0.y = bp[LS + 0];
        b1.x = bp[16]; b1.y = bp[LS + 16];
        b2.x = bp[32]; b2.y = bp[LS + 32];
        b3.x = bp[48]; b3.y = bp[LS + 48];
        acc0 = __builtin_amdgcn_wmma_f32_16x16x4_f32(false, a, false, b0, (short)0, acc0, false, false);
        acc1 = __builtin_amdgcn_wmma_f32_16x16x4_f32(false, a, false, b1, (short)0, acc1, false, false);
        acc2 = __builtin_amdgcn_wmma_f32_16x16x4_f32(false, a, false, b2, (short)0, acc2, false, false);
        acc3 = __builtin_amdgcn_wmma_f32_16x16x4_f32(false, a, false, b3, (short)0, acc3, false, false);
    }

    // C/D: VGPR r -> (M = row0 + r + 8*half, N = 16*ct + mn). +I only hits tile ct==wv.
#define STORE_TILE(CT, ACC)                                                    \
    {                                                                          \
        _Pragma("unroll")                                                      \
        for (int r = 0; r < 8; ++r) {                                          \
            float v = ACC[r] * scale;                                          \
            if (addI && (CT) == wv && (r + 8 * half) == mn) v += 1.0f;         \
            Dst[(row0 + r + 8 * half) * LS + ((CT) << 4) + mn] = v;            \
        }                                                                      \
    }
    STORE_TILE(0, acc0)
    STORE_TILE(1, acc1)
    STORE_TILE(2, acc2)
    STORE_TILE(3, acc3)
#undef STORE_TILE
}

// exp(0.5*(X+X^T)) per 64x64 batch element, via scaling-and-squaring + Horner Taylor.
__global__ __launch_bounds__(THREADS) void spd_expm_kernel(const float* __restrict__ X,
                                                           float* __restrict__ Out)
{
    __shared__ __align__(16) float Abuf[64 * LS];     // scaled symmetric input
    __shared__ __align__(16) float Pbuf[2][64 * LS];  // double-buffered accumulator
    __shared__ float red[64];

    const int    tid  = threadIdx.x;
    const int    lane = tid & 31;
    const int    wv   = tid >> 5;
    const size_t base = (size_t)blockIdx.x * 4096;

    // --- stage X coalesced into Pbuf[1] (dead until first Horner matmul) ---
    float* Xl = Pbuf[1];
    for (int idx = tid; idx < 4096; idx += THREADS) {
        const int r = idx >> 6, c = idx & 63;
        Xl[r * LS + c] = X[base + idx];
    }
    __syncthreads();

    // --- symmetrize from LDS: S = 0.5*(X + X^T) ---
    for (int idx = tid; idx < 4096; idx += THREADS) {
        const int r = idx >> 6, c = idx & 63;
        Abuf[r * LS + c] = 0.5f * (Xl[r * LS + c] + Xl[c * LS + r]);
    }
    __syncthreads();

    // --- inf-norm (max abs row sum) to pick the scaling exponent ---
    if (tid < 64) {
        float s = 0.0f;
        for (int c = 0; c < 64; ++c) s += fabsf(Abuf[tid * LS + c]);
        red[tid] = s;
    }
    __syncthreads();
    if (tid == 0) {
        float m = red[0];
        for (int i = 1; i < 64; ++i) m = fmaxf(m, red[i]);
        red[0] = m;
    }
    __syncthreads();

    const float norm = red[0];
    int sexp = 0;
    if (norm > 0.5f) sexp = (int)ceilf(log2f(norm) + 1.0f);  // norm/2^s <= 0.5
    if (sexp < 0)  sexp = 0;
    if (sexp > 15) sexp = 15;
    const float sc = exp2f(-(float)sexp);

    // --- scale A in place; init Horner: P = I + A/N ---
    for (int idx = tid; idx < 4096; idx += THREADS) {
        const int r = idx >> 6, c = idx & 63;
        const float a = Abuf[r * LS + c] * sc;
        Abuf[r * LS + c] = a;
        Pbuf[0][r * LS + c] = a * (1.0f / (float)TAYLOR_N) + ((r == c) ? 1.0f : 0.0f);
    }
    __syncthreads();

    // --- Horner: P <- I + (A*P)/k, k = N-1 .. 1 ---
    int cur = 0;
    for (int k = TAYLOR_N - 1; k >= 1; --k) {
        const float invk = __builtin_amdgcn_rcpf((float)k);   // 1-ulp v_rcp_f32
        band_matmul(Abuf, Pbuf[cur], Pbuf[cur ^ 1], wv, lane, invk, true);
        __syncthreads();
        cur ^= 1;
    }

    // --- squaring phase: P <- P*P, s times ---
    for (int t = 0; t < sexp; ++t) {
        band_matmul(Pbuf[cur], Pbuf[cur], Pbuf[cur ^ 1], wv, lane, 1.0f, false);
        __syncthreads();
        cur ^= 1;
    }

    // --- write result (coalesced) ---
    for (int idx = tid; idx < 4096; idx += THREADS) {
        const int r = idx >> 6, c = idx & 63;
        Out[base + idx] = Pbuf[cur][r * LS + c];
    }
}

extern "C" void kernel_launch(void* const* d_in, const int* in_sizes, int n_in,
                              void* d_out, int out_size, void* d_ws, size_t ws_size,
                              hipStream_t stream) {
    (void)n_in; (void)out_size; (void)d_ws; (void)ws_size;
    const float* X   = (const float*)d_in[0];
    float*       out = (float*)d_out;
    const int nBatch = in_sizes[0] / 4096;   // 8192 matrices of 64x64
    spd_expm_kernel<<<nBatch, THREADS, 0, stream>>>(X, out);
}